// GraphSAGE_47725676593431
// MI455X (gfx1250) — compile-verified
//
#include <hip/hip_runtime.h>

typedef __attribute__((ext_vector_type(2))) float v2f;
typedef __attribute__((ext_vector_type(8))) float v8f;

#define D 128

// ---------------------------------------------------------------------------
// Fused degree-count + neighbor scatter-add.
// One wave (32 lanes) per edge; each lane moves one float4 (16B) => the wave
// moves the whole 512B feature row coalesced, then does 4 f32 global atomics.
// ---------------------------------------------------------------------------
__global__ void __launch_bounds__(256)
sage_scatter(const float* __restrict__ feat, const int* __restrict__ src,
             const int* __restrict__ dst, float* __restrict__ agg,
             float* __restrict__ cnt, int nEdges, int addCnt)
{
    int tid  = blockIdx.x * blockDim.x + threadIdx.x;
    int e    = tid >> 5;
    int lane = tid & 31;
    if (e >= nEdges) return;
    int s = src[e];
    int d = dst[e];
    const float4* frow = (const float4*)(feat + (size_t)s * D);
    float4 v = frow[lane];
    float* arow = agg + (size_t)d * D + lane * 4;
    atomicAdd(arow + 0, v.x);
    atomicAdd(arow + 1, v.y);
    atomicAdd(arow + 2, v.z);
    atomicAdd(arow + 3, v.w);
    if (addCnt && lane == 0) atomicAdd(cnt + d, 1.0f);
}

// ---------------------------------------------------------------------------
// Stage a 128x128 row-major weight matrix into LDS as interleaved row pairs:
//   sW2[p*128 + n] = { W[2p][n], W[2p+1][n] },  p in [0,64)
// so a WMMA B-fragment {W[k0][n], W[k0+1][n]} (k0 even) is ONE contiguous
// 8-byte LDS load landing directly in an even-aligned VGPR pair.
// ---------------------------------------------------------------------------
__device__ __forceinline__ void stage_w_pairs(const float* __restrict__ W,
                                              float2* sW2, int t)
{
    // 64 row-pairs * 32 float4 columns = 2048 chunks, 8 per thread
    for (int i = t; i < 64 * 32; i += 256) {
        int p  = i >> 5;   // row pair index
        int c4 = i & 31;   // float4 column
        float4 r0 = ((const float4*)(W + (size_t)(2 * p) * D))[c4];
        float4 r1 = ((const float4*)(W + (size_t)(2 * p + 1) * D))[c4];
        float2* dstp = sW2 + p * D + c4 * 4;
        dstp[0] = make_float2(r0.x, r1.x);
        dstp[1] = make_float2(r0.y, r1.y);
        dstp[2] = make_float2(r0.z, r1.z);
        dstp[3] = make_float2(r0.w, r1.w);
    }
}

// ---------------------------------------------------------------------------
// Fused dual-GEMM epilogue:  out = act( (agg*inv(cnt)) @ Wl + Ax @ Wr + b )
// 256 threads = 8 waves; each wave owns a 16x128 output strip (wg = 128 rows).
// Two passes (Wl then Wr) accumulate into the same fp32 WMMA accumulators via
// V_WMMA_F32_16X16X4_F32.
//
// 16x16x4 f32 fragment mapping (wave32):
//   A: a[v] = A[m][k],  m = lane&15, k = 2*(lane>>4) + v
//   B: b[v] = B[k][n],  n = lane&15, k = 2*(lane>>4) + v
//   C/D: c[r] = C[r + 8*(lane>>4)][lane&15]
// ---------------------------------------------------------------------------
__global__ void __launch_bounds__(256)
sage_gemm(const float* __restrict__ Aagg, const float* __restrict__ Ax,
          const float* __restrict__ cnt, const float* __restrict__ Wl,
          const float* __restrict__ Wr, const float* __restrict__ bias,
          float* __restrict__ out, int nRows, int doRelu)
{
    __shared__ float2 sW2[64 * D];   // 64 KB, interleaved row pairs

    int t    = threadIdx.x;
    int wave = t >> 5;
    int lane = t & 31;
    int rowBase = (blockIdx.x * 8 + wave) * 16;
    bool active = (rowBase + 16) <= nRows;   // wave-uniform => EXEC stays full
    int mn    = lane & 15;                   // m for A, n for B/C
    int khalf = (lane >> 4) * 2;             // K sub-offset per lane half
    int phalf = (lane >> 4);                 // row-pair sub-offset

    const v2f* sB = (const v2f*)sW2;

    v8f acc[8];
    v8f zero = {0.f, 0.f, 0.f, 0.f, 0.f, 0.f, 0.f, 0.f};
#pragma unroll
    for (int j = 0; j < 8; ++j) acc[j] = zero;

    // ---- stage Wl ----
    stage_w_pairs(Wl, sW2, t);
    __syncthreads();

    // ---- pass 1: acc += (agg * 1/max(cnt,1)) @ Wl ----
    if (active) {
        float c   = cnt[rowBase + mn];
        float inv = 1.0f / fmaxf(c, 1.0f);
        const float* arow = Aagg + (size_t)(rowBase + mn) * D;
        for (int kt = 0; kt < D; kt += 4) {
            v2f a = *(const v2f*)(arow + kt + khalf);   // global_load_b64
            a.x *= inv;
            a.y *= inv;
            const v2f* brow = sB + ((kt >> 1) + phalf) * D + mn;
#pragma unroll
            for (int j = 0; j < 8; ++j) {
                v2f b = brow[j * 16];                   // ds_load_b64
                acc[j] = __builtin_amdgcn_wmma_f32_16x16x4_f32(
                    false, a, false, b, (short)0, acc[j], false, false);
            }
        }
    }
    __syncthreads();

    // ---- stage Wr ----
    stage_w_pairs(Wr, sW2, t);
    __syncthreads();

    // ---- pass 2: acc += Ax @ Wr ; epilogue bias (+relu) + store ----
    if (active) {
        const float* xrow = Ax + (size_t)(rowBase + mn) * D;
        for (int kt = 0; kt < D; kt += 4) {
            v2f a = *(const v2f*)(xrow + kt + khalf);
            const v2f* brow = sB + ((kt >> 1) + phalf) * D + mn;
#pragma unroll
            for (int j = 0; j < 8; ++j) {
                v2f b = brow[j * 16];
                acc[j] = __builtin_amdgcn_wmma_f32_16x16x4_f32(
                    false, a, false, b, (short)0, acc[j], false, false);
            }
        }

        int rowOff = (lane >> 4) * 8;   // lanes 16..31 hold M = 8..15
#pragma unroll
        for (int j = 0; j < 8; ++j) {
            float bv = bias[j * 16 + mn];
#pragma unroll
            for (int r = 0; r < 8; ++r) {
                float val = acc[j][r] + bv;
                if (doRelu) val = fmaxf(val, 0.0f);
                out[(size_t)(rowBase + rowOff + r) * D + j * 16 + mn] = val;
            }
        }
    }
}

// ---------------------------------------------------------------------------
// Host-side launch. Inputs (dict order): x, edge_index, W1l, W1r, b1, W2l,
// W2r, b2. edge_index is int32 (JAX x64 disabled). Workspace: agg (N*128 f32)
// + cnt (N f32) ~ 51.6 MB. Layer-1 activations live in d_out.
// ---------------------------------------------------------------------------
extern "C" void kernel_launch(void* const* d_in, const int* in_sizes, int n_in,
                              void* d_out, int out_size, void* d_ws, size_t ws_size,
                              hipStream_t stream)
{
    const float* x   = (const float*)d_in[0];
    const int*   ei  = (const int*)d_in[1];
    const float* W1l = (const float*)d_in[2];
    const float* W1r = (const float*)d_in[3];
    const float* b1  = (const float*)d_in[4];
    const float* W2l = (const float*)d_in[5];
    const float* W2r = (const float*)d_in[6];
    const float* b2  = (const float*)d_in[7];

    int N = in_sizes[0] / D;
    int E = in_sizes[1] / 2;
    const int* src = ei;
    const int* dst = ei + E;

    float* agg = (float*)d_ws;
    float* cnt = agg + (size_t)N * D;
    float* h   = (float*)d_out;   // layer-1 output
    float* out = (float*)d_out;

    hipMemsetAsync(agg, 0, (size_t)N * D * sizeof(float), stream);
    hipMemsetAsync(cnt, 0, (size_t)N * sizeof(float), stream);

    int scatterBlocks = (E + 7) / 8;        // 8 edges (waves) per block
    int strips        = (N + 15) / 16;
    int gemmBlocks    = (strips + 7) / 8;   // 8 strips per block

    // Layer 1
    sage_scatter<<<scatterBlocks, 256, 0, stream>>>(x, src, dst, agg, cnt, E, 1);
    sage_gemm<<<gemmBlocks, 256, 0, stream>>>(agg, x, cnt, W1l, W1r, b1, h, N, 1);

    // Layer 2
    hipMemsetAsync(agg, 0, (size_t)N * D * sizeof(float), stream);
    sage_scatter<<<scatterBlocks, 256, 0, stream>>>(h, src, dst, agg, cnt, E, 0);
    sage_gemm<<<gemmBlocks, 256, 0, stream>>>(agg, h, cnt, W2l, W2r, b2, out, N, 0);
}